// MaskedSelfAttention_79474074845767
// MI455X (gfx1250) — compile-verified
//
#include <hip/hip_runtime.h>

// MaskedSelfAttention, MI455X (gfx1250), all-fp32 path via V_WMMA_F32_16X16X4_F32.
// B=4, S=256, HID=512, NH=8, D=64. scale = 1/sqrt(64) = 0.125.
// GEMM kernels use 32x32 tiles per wave (2x2 WMMA fragments) for 2x A/B reuse.
//
// Workspace layout (bytes):           size
//   Q0    [B,H,S,D] f32               2 MB
//   K0    [B,H,S,D] f32               2 MB
//   V0    [B,H,S,D] f32               2 MB
//   QROW  [B,H,S,D] f32               2 MB
//   QRDOT [B,H,S,8] f32               256 KB
//   CNT   [B,S,S] uint4 (8 x u16)     4 MB
//   SCORES[B,H,S,S] f32               8 MB
// total ~20.25 MB

constexpr int kB   = 4;
constexpr int kS   = 256;
constexpr int kHID = 512;
constexpr int kNH  = 8;
constexpr int kD   = 64;

typedef float v2f __attribute__((ext_vector_type(2)));
typedef float v8f __attribute__((ext_vector_type(8)));

#define WMMA_F32(a, b, c) \
  __builtin_amdgcn_wmma_f32_16x16x4_f32(false, (a), false, (b), (short)0, (c), false, false)

__device__ __forceinline__ float wave_sum(float v) {
#pragma unroll
  for (int o = 16; o > 0; o >>= 1) v += __shfl_xor(v, o, 32);
  return v;
}
__device__ __forceinline__ float wave_max(float v) {
#pragma unroll
  for (int o = 16; o > 0; o >>= 1) v = fmaxf(v, __shfl_xor(v, o, 32));
  return v;
}

// ---------------------------------------------------------------------------
// K1: Out[b,h,s,d] = (X @ W)[row,col] + bias[col], row=b*S+s, col=h*D+d.
// One wave -> 32x32 tile (2x2 fragments). M=1024, N=512, K=512.
// 512 wave-tiles -> 64 blocks of 8 waves.
// ---------------------------------------------------------------------------
__global__ __launch_bounds__(256) void proj_gemm_kernel(
    const float* __restrict__ X, const float* __restrict__ W,
    const float* __restrict__ bias, float* __restrict__ Out) {
  const int lane = threadIdx.x & 31;
  const int wave = threadIdx.x >> 5;
  const int lm   = lane & 15;      // within-fragment row/col index
  const int half = lane >> 4;      // lane half selects K pair
  const int tile = blockIdx.x * 8 + wave;       // [0, 512)
  const int tilesN = kHID / 32;                 // 16
  const int m0 = (tile / tilesN) * 32;
  const int n0 = (tile % tilesN) * 32;

  const float* __restrict__ A0 = X + (size_t)(m0 + lm) * kHID;
  const float* __restrict__ A1 = X + (size_t)(m0 + 16 + lm) * kHID;
  const float* __restrict__ B0 = W + n0 + lm;
  const float* __restrict__ B1 = W + n0 + 16 + lm;

  v8f c00 = {}, c01 = {}, c10 = {}, c11 = {};
  for (int kk = 0; kk < kHID; kk += 4) {
    const int ka = kk + 2 * half;
    v2f a0; a0.x = A0[ka]; a0.y = A0[ka + 1];
    v2f a1; a1.x = A1[ka]; a1.y = A1[ka + 1];
    v2f b0; b0.x = B0[(size_t)ka * kHID]; b0.y = B0[(size_t)(ka + 1) * kHID];
    v2f b1; b1.x = B1[(size_t)ka * kHID]; b1.y = B1[(size_t)(ka + 1) * kHID];
    c00 = WMMA_F32(a0, b0, c00);
    c01 = WMMA_F32(a0, b1, c01);
    c10 = WMMA_F32(a1, b0, c10);
    c11 = WMMA_F32(a1, b1, c11);
  }

#pragma unroll
  for (int ni = 0; ni < 2; ++ni) {
    const int col = n0 + ni * 16 + lm;
    const float bvv = bias[col];
    const int h = col >> 6;          // col / D
    const int d = col & (kD - 1);
#pragma unroll
    for (int mi = 0; mi < 2; ++mi) {
      const v8f& c = (mi == 0) ? (ni == 0 ? c00 : c01) : (ni == 0 ? c10 : c11);
#pragma unroll
      for (int r = 0; r < 8; ++r) {
        const int row = m0 + mi * 16 + r + 8 * half;
        const int b_ = row >> 8;     // row / S
        const int s  = row & (kS - 1);
        Out[((size_t)((b_ * kNH + h) * kS + s) << 6) + d] = c[r] + bvv;
      }
    }
  }
}

// ---------------------------------------------------------------------------
// K2: running 8-bin histogram along i:  CNT[b,i,j] = counts of edge_type[b,t,j]
// for t<=i, packed as 8 x u16 in a uint4. One thread per (b,j).
// ---------------------------------------------------------------------------
__global__ __launch_bounds__(256) void cnt_kernel(
    const int* __restrict__ edge_type, uint4* __restrict__ CNT) {
  const int tid = blockIdx.x * blockDim.x + threadIdx.x;   // B*S threads
  const int b_ = tid >> 8;
  const int j  = tid & (kS - 1);
  const int*  __restrict__ ep   = edge_type + (size_t)b_ * kS * kS + j;
  uint4*      __restrict__ outp = CNT       + (size_t)b_ * kS * kS + j;
  uint4 cw = {0u, 0u, 0u, 0u};
  for (int i = 0; i < kS; ++i) {
    const int e = ep[(size_t)i * kS];
    const unsigned inc = 1u << (16 * (e & 1));
    const int w = e >> 1;
    cw.x += (w == 0) ? inc : 0u;
    cw.y += (w == 1) ? inc : 0u;
    cw.z += (w == 2) ? inc : 0u;
    cw.w += (w == 3) ? inc : 0u;
    outp[(size_t)i * kS] = cw;
  }
}

// ---------------------------------------------------------------------------
// K3: qrow = Q0 + diagC, qr_dot[e] = qrow . rel_table[e, hD:hD+D].
// One wave per (b,h,i) row; lane owns d = lane and d = lane+32.
// ---------------------------------------------------------------------------
__global__ __launch_bounds__(256) void qrow_kernel(
    const float* __restrict__ Q0, const float* __restrict__ rel,
    const uint4* __restrict__ CNT, float* __restrict__ QROW,
    float* __restrict__ QRDOT) {
  const int lane = threadIdx.x & 31;
  const int wave = threadIdx.x >> 5;
  const int row = blockIdx.x * 8 + wave;   // [B*H*S)
  const int i  = row & (kS - 1);
  const int bh = row >> 8;
  const int h  = bh & (kNH - 1);
  const int b_ = bh >> 3;

  const uint4 cw = CNT[((size_t)(b_ * kS + i)) * kS + i];
  float cnt[8];
  cnt[0] = (float)(cw.x & 0xffffu); cnt[1] = (float)(cw.x >> 16);
  cnt[2] = (float)(cw.y & 0xffffu); cnt[3] = (float)(cw.y >> 16);
  cnt[4] = (float)(cw.z & 0xffffu); cnt[5] = (float)(cw.z >> 16);
  cnt[6] = (float)(cw.w & 0xffffu); cnt[7] = (float)(cw.w >> 16);

  const float* __restrict__ relh = rel + h * kD;          // rel[e*HID + h*D + d]
  const float* __restrict__ q    = Q0 + (size_t)row * kD;

  float dot[8] = {0.f, 0.f, 0.f, 0.f, 0.f, 0.f, 0.f, 0.f};
#pragma unroll
  for (int t = 0; t < 2; ++t) {
    const int d = lane + 32 * t;
    float rv[8];
#pragma unroll
    for (int e = 0; e < 8; ++e) rv[e] = relh[e * kHID + d];
    float v = q[d];
#pragma unroll
    for (int e = 0; e < 8; ++e) v = fmaf(cnt[e], rv[e], v);
    QROW[(size_t)row * kD + d] = v;
#pragma unroll
    for (int e = 0; e < 8; ++e) dot[e] = fmaf(v, rv[e], dot[e]);
  }
#pragma unroll
  for (int e = 0; e < 8; ++e) {
    const float s = wave_sum(dot[e]);
    if (lane == 0) QRDOT[(size_t)row * 8 + e] = s;
  }
}

// ---------------------------------------------------------------------------
// K4: scores[b,h,i,j] = ((qrow . K0) + sum_e cnt*qr_dot) * 0.125, masked.
// One wave -> 32x32 (i,j) tile (2x2 fragments), K-loop over D=64.
// 64 tiles per (b,h) * 32 = 2048 wave-tiles -> 256 blocks.
// ---------------------------------------------------------------------------
__global__ __launch_bounds__(256) void scores_kernel(
    const float* __restrict__ QROW, const float* __restrict__ K0,
    const float* __restrict__ QRDOT, const uint4* __restrict__ CNT,
    const int* __restrict__ trans_mask, float* __restrict__ SCORES) {
  const int lane = threadIdx.x & 31;
  const int wave = threadIdx.x >> 5;
  const int lm   = lane & 15;
  const int half = lane >> 4;
  const int t   = blockIdx.x * 8 + wave;    // [0, 2048)
  const int bh  = t >> 6;
  const int rem = t & 63;
  const int i0  = (rem >> 3) * 32;
  const int j0  = (rem & 7) * 32;
  const int b_  = bh >> 3;

  const float* __restrict__ A0 = QROW + ((size_t)bh * kS + i0 + lm) * kD;
  const float* __restrict__ A1 = QROW + ((size_t)bh * kS + i0 + 16 + lm) * kD;
  const float* __restrict__ B0 = K0   + ((size_t)bh * kS + j0 + lm) * kD;
  const float* __restrict__ B1 = K0   + ((size_t)bh * kS + j0 + 16 + lm) * kD;

  v8f c00 = {}, c01 = {}, c10 = {}, c11 = {};
#pragma unroll
  for (int kk = 0; kk < kD; kk += 4) {
    const int ka = kk + 2 * half;
    v2f a0; a0.x = A0[ka]; a0.y = A0[ka + 1];
    v2f a1; a1.x = A1[ka]; a1.y = A1[ka + 1];
    v2f b0; b0.x = B0[ka]; b0.y = B0[ka + 1];   // B[k][n] = K0[j=n][k]
    v2f b1; b1.x = B1[ka]; b1.y = B1[ka + 1];
    c00 = WMMA_F32(a0, b0, c00);
    c01 = WMMA_F32(a0, b1, c01);
    c10 = WMMA_F32(a1, b0, c10);
    c11 = WMMA_F32(a1, b1, c11);
  }

#pragma unroll
  for (int mi = 0; mi < 2; ++mi) {
#pragma unroll
    for (int r = 0; r < 8; ++r) {
      const int i = i0 + mi * 16 + r + 8 * half;
      const float4* __restrict__ qd4 =
          (const float4*)(QRDOT + ((size_t)bh * kS + i) * 8);
      const float4 qa = qd4[0];
      const float4 qb = qd4[1];
#pragma unroll
      for (int ni = 0; ni < 2; ++ni) {
        const int j = j0 + ni * 16 + lm;
        const v8f& c = (mi == 0) ? (ni == 0 ? c00 : c01) : (ni == 0 ? c10 : c11);
        const uint4 cw = CNT[((size_t)(b_ * kS + i)) * kS + j];
        float s2 = 0.f;
        s2 = fmaf((float)(cw.x & 0xffffu), qa.x, s2);
        s2 = fmaf((float)(cw.x >> 16),     qa.y, s2);
        s2 = fmaf((float)(cw.y & 0xffffu), qa.z, s2);
        s2 = fmaf((float)(cw.y >> 16),     qa.w, s2);
        s2 = fmaf((float)(cw.z & 0xffffu), qb.x, s2);
        s2 = fmaf((float)(cw.z >> 16),     qb.y, s2);
        s2 = fmaf((float)(cw.w & 0xffffu), qb.z, s2);
        s2 = fmaf((float)(cw.w >> 16),     qb.w, s2);
        float sc = (c[r] + s2) * 0.125f;
        if (trans_mask[((size_t)(b_ * kS + i)) * kS + j] == 0) sc = -1e9f;
        SCORES[((size_t)bh * kS + i) * kS + j] = sc;
      }
    }
  }
}

// ---------------------------------------------------------------------------
// K5: in-place row softmax over j (256 elems). One wave per (b,h,i) row.
// ---------------------------------------------------------------------------
__global__ __launch_bounds__(256) void softmax_kernel(float* __restrict__ SCORES) {
  const int lane = threadIdx.x & 31;
  const int wave = threadIdx.x >> 5;
  const int row = blockIdx.x * 8 + wave;     // B*H*S rows
  float* __restrict__ p = SCORES + (size_t)row * kS;
  float v[8];
  float mx = -3.4e38f;
#pragma unroll
  for (int t = 0; t < 8; ++t) { v[t] = p[t * 32 + lane]; mx = fmaxf(mx, v[t]); }
  mx = wave_max(mx);
  float sum = 0.f;
#pragma unroll
  for (int t = 0; t < 8; ++t) { v[t] = __expf(v[t] - mx); sum += v[t]; }
  sum = wave_sum(sum);
  const float inv = 1.f / sum;
#pragma unroll
  for (int t = 0; t < 8; ++t) p[t * 32 + lane] = v[t] * inv;
}

// ---------------------------------------------------------------------------
// K6: ctx = probs @ V0 per (b,h); write out[b,s,h*D+d].
// One wave -> 32x32 tile (2x2 fragments). 16 tiles/(b,h) * 32 = 512 -> 64 blocks.
// ---------------------------------------------------------------------------
__global__ __launch_bounds__(256) void ctx_kernel(
    const float* __restrict__ PROBS, const float* __restrict__ V0,
    float* __restrict__ Out) {
  const int lane = threadIdx.x & 31;
  const int wave = threadIdx.x >> 5;
  const int lm   = lane & 15;
  const int half = lane >> 4;
  const int t   = blockIdx.x * 8 + wave;   // [0, 512)
  const int bh  = t >> 4;
  const int rem = t & 15;
  const int i0  = (rem >> 1) * 32;
  const int d0  = (rem & 1) * 32;
  const int b_  = bh >> 3;
  const int h   = bh & (kNH - 1);

  const float* __restrict__ A0 = PROBS + (size_t)bh * kS * kS + (size_t)(i0 + lm) * kS;
  const float* __restrict__ A1 = PROBS + (size_t)bh * kS * kS + (size_t)(i0 + 16 + lm) * kS;
  const float* __restrict__ B0 = V0 + (size_t)bh * kS * kD + d0 + lm;
  const float* __restrict__ B1 = V0 + (size_t)bh * kS * kD + d0 + 16 + lm;

  v8f c00 = {}, c01 = {}, c10 = {}, c11 = {};
  for (int kk = 0; kk < kS; kk += 4) {
    const int ka = kk + 2 * half;
    v2f a0; a0.x = A0[ka]; a0.y = A0[ka + 1];
    v2f a1; a1.x = A1[ka]; a1.y = A1[ka + 1];
    v2f b0; b0.x = B0[(size_t)ka * kD]; b0.y = B0[(size_t)(ka + 1) * kD];
    v2f b1; b1.x = B1[(size_t)ka * kD]; b1.y = B1[(size_t)(ka + 1) * kD];
    c00 = WMMA_F32(a0, b0, c00);
    c01 = WMMA_F32(a0, b1, c01);
    c10 = WMMA_F32(a1, b0, c10);
    c11 = WMMA_F32(a1, b1, c11);
  }

#pragma unroll
  for (int mi = 0; mi < 2; ++mi) {
#pragma unroll
    for (int ni = 0; ni < 2; ++ni) {
      const v8f& c = (mi == 0) ? (ni == 0 ? c00 : c01) : (ni == 0 ? c10 : c11);
      const int d = d0 + ni * 16 + lm;
#pragma unroll
      for (int r = 0; r < 8; ++r) {
        const int i = i0 + mi * 16 + r + 8 * half;
        Out[((size_t)(b_ * kS + i)) * kHID + h * kD + d] = c[r];
      }
    }
  }
}

// ---------------------------------------------------------------------------
extern "C" void kernel_launch(void* const* d_in, const int* in_sizes, int n_in,
                              void* d_out, int out_size, void* d_ws, size_t ws_size,
                              hipStream_t stream) {
  const float* q_h  = (const float*)d_in[0];
  const float* k_h  = (const float*)d_in[1];
  const float* v_h  = (const float*)d_in[2];
  const float* Wq   = (const float*)d_in[3];
  const float* bq   = (const float*)d_in[4];
  const float* Wk   = (const float*)d_in[5];
  const float* bk   = (const float*)d_in[6];
  const float* Wv   = (const float*)d_in[7];
  const float* bv   = (const float*)d_in[8];
  const float* rel  = (const float*)d_in[9];
  const int*   edge = (const int*)d_in[10];
  const int*   mask = (const int*)d_in[11];
  float* out = (float*)d_out;

  char* ws = (char*)d_ws;
  const size_t nBHSD = (size_t)kB * kNH * kS * kD;    // 524288
  float* Q0     = (float*)(ws);
  float* K0     = (float*)(ws + 4 * nBHSD);
  float* V0     = (float*)(ws + 8 * nBHSD);
  float* QROW   = (float*)(ws + 12 * nBHSD);
  float* QRDOT  = (float*)(ws + 16 * nBHSD);
  uint4* CNT    = (uint4*)(ws + 16 * nBHSD + 4 * (size_t)kB * kNH * kS * 8);
  float* SCORES = (float*)((char*)CNT + sizeof(uint4) * (size_t)kB * kS * kS);

  // K1: QKV projections (512 wave-tiles each -> 64 blocks of 8 waves)
  proj_gemm_kernel<<<64, 256, 0, stream>>>(q_h, Wq, bq, Q0);
  proj_gemm_kernel<<<64, 256, 0, stream>>>(k_h, Wk, bk, K0);
  proj_gemm_kernel<<<64, 256, 0, stream>>>(v_h, Wv, bv, V0);

  // K2: running histograms (B*S = 1024 threads)
  cnt_kernel<<<4, 256, 0, stream>>>(edge, CNT);

  // K3: qrow + qr_dot (8192 rows, wave per row)
  qrow_kernel<<<1024, 256, 0, stream>>>(Q0, rel, CNT, QROW, QRDOT);

  // K4: scores (2048 wave-tiles)
  scores_kernel<<<256, 256, 0, stream>>>(QROW, K0, QRDOT, CNT, mask, SCORES);

  // K5: softmax (8192 rows)
  softmax_kernel<<<1024, 256, 0, stream>>>(SCORES);

  // K6: ctx (512 wave-tiles)
  ctx_kernel<<<64, 256, 0, stream>>>(SCORES, V0, out);
}